// SS2D_4982162063930
// MI455X (gfx1250) — compile-verified
//
#include <hip/hip_runtime.h>
#include <math.h>

// ---------------------------------------------------------------------------
// SS2D (VMamba) block for MI455X / gfx1250, wave32, fp32 WMMA (16x16x4).
// All GEMMs use V_WMMA_F32_16X16X4_F32 (exact fp32, matches reference math).
// Guard-free, software-pipelined mainloop: next step's fragments are loaded
// before the current step's wmma pair issues, so VMEM latency overlaps with
// matrix issue instead of s_wait_loadcnt 0 before every wmma.
// ---------------------------------------------------------------------------

typedef float v2f __attribute__((ext_vector_type(2)));
typedef float v8f __attribute__((ext_vector_type(8)));

#define BB 4
#define HH 48
#define WW 48
#define DMm 192
#define DIc 192
#define KK 4
#define NN 16
#define RR 12
#define LL (HH * WW)          // 2304
#define CPROJ (RR + 2 * NN)   // 44
#define CPAD 64               // x_proj weight rows padded to 64

__device__ __forceinline__ float siluf(float x)     { return x / (1.f + __expf(-x)); }
__device__ __forceinline__ float sigm(float x)      { return 1.f / (1.f + __expf(-x)); }
__device__ __forceinline__ float softplusf_(float x){ return (x > 20.f) ? x : log1pf(__expf(x)); }
__device__ __forceinline__ float geluf(float x)     { return 0.5f * x * (1.f + erff(x * 0.70710678118654752440f)); }

// ---------------------------------------------------------------------------
// WMMA mainloop, NT adjacent 16-wide col-tiles per wave (one shared A frag).
// O[row,col] = sum_k X[row,k] * W[col,k];  X row-major (ldx), W row-major (ldw).
// Kdim must be a multiple of 4; all fragment loads are unconditionally valid.
// A-fragment (16x4 f32):  lane = (k>=2)*16 + m, vgpr = k&1
// B-fragment (4x16 f32):  lane = (k>=2)*16 + n, vgpr = k&1
// C/D (16x16 f32): vgpr v -> m = v + (lane>>4)*8, n = lane&15
// 1-deep software pipeline: loads for step k+1 issue before wmma of step k.
// ---------------------------------------------------------------------------
template <int NT>
__device__ __forceinline__ void wmma_xwT_multi(const float* __restrict__ X,
                                               const float* __restrict__ W,
                                               int row0, int col0,
                                               int Kdim, int ldx, int ldw,
                                               v8f acc[NT])
{
    const int lane = threadIdx.x & 31;
    const int m    = lane & 15;
    const int kh   = lane >> 4;           // 0: K{0,1}, 1: K{2,3}
    const float* __restrict__ ap = X + (size_t)(row0 + m) * ldx + 2 * kh;
    const float* __restrict__ bp[NT];
#pragma unroll
    for (int t = 0; t < NT; ++t) {
        bp[t] = W + (size_t)(col0 + t * 16 + m) * ldw + 2 * kh;
        acc[t] = {};
    }
    // pipeline prologue: stage-0 fragment loads
    v2f a_cur = *(const v2f*)(ap);
    v2f b_cur[NT];
#pragma unroll
    for (int t = 0; t < NT; ++t) b_cur[t] = *(const v2f*)(bp[t]);

    for (int k0 = 4; k0 < Kdim; k0 += 4) {
        // issue next step's loads before consuming current fragments
        v2f a_nxt = *(const v2f*)(ap + k0);
        v2f b_nxt[NT];
#pragma unroll
        for (int t = 0; t < NT; ++t) b_nxt[t] = *(const v2f*)(bp[t] + k0);
#pragma unroll
        for (int t = 0; t < NT; ++t)
            acc[t] = __builtin_amdgcn_wmma_f32_16x16x4_f32(false, a_cur, false, b_cur[t],
                                                           (short)0, acc[t], false, false);
        a_cur = a_nxt;
#pragma unroll
        for (int t = 0; t < NT; ++t) b_cur[t] = b_nxt[t];
    }
    // epilogue step
#pragma unroll
    for (int t = 0; t < NT; ++t)
        acc[t] = __builtin_amdgcn_wmma_f32_16x16x4_f32(false, a_cur, false, b_cur[t],
                                                       (short)0, acc[t], false, false);
}

// ----------------------- 0) pad x_proj weights to CPAD rows ----------------
__global__ void k_padw(const float* __restrict__ xpw, float* __restrict__ wpad)
{
    int k = blockIdx.x, c = blockIdx.y, d = threadIdx.x;
    wpad[((size_t)k * CPAD + c) * DIc + d] =
        (c < CPROJ) ? xpw[((size_t)k * CPROJ + c) * DIc + d] : 0.f;
}

// ----------------------- 1) in_proj: xz = x @ Win^T ------------------------
// x: (B*L, DM) row-major, Win: (2*DI, DM). cols [0,DI) -> xc_lin, [DI,2DI) -> silu -> z
__global__ void k_inproj(const float* __restrict__ x, const float* __restrict__ w,
                         float* __restrict__ xc_lin, float* __restrict__ z_silu)
{
    const int row0 = blockIdx.y * 16, col0 = blockIdx.x * 32;
    v8f acc[2];
    wmma_xwT_multi<2>(x, w, row0, col0, DMm, DMm, DMm, acc);
    const int lane = threadIdx.x & 31;
    const int nn = lane & 15, mh = (lane >> 4) * 8;
#pragma unroll
    for (int t = 0; t < 2; ++t)
#pragma unroll
        for (int v = 0; v < 8; ++v) {
            int r = row0 + mh + v;
            int c = col0 + t * 16 + nn;
            float val = acc[t][v];
            if (c < DIc) xc_lin[(size_t)r * DIc + c] = val;
            else         z_silu[(size_t)r * DIc + (c - DIc)] = siluf(val);
        }
}

// --------------- 2) depthwise 3x3 conv (SAME, cross-corr) + silu -----------
// layouts: (B,H,W,DI) channel-contiguous. One block per pixel, 192 threads=channels.
__global__ void k_dwconv(const float* __restrict__ xc_lin, const float* __restrict__ cw,
                         const float* __restrict__ cb, float* __restrict__ xc_act)
{
    int pix = blockIdx.x;
    int b = pix / LL, rem = pix % LL;
    int h = rem / WW, w = rem % WW;
    int c = threadIdx.x;
    float s = cb[c];
#pragma unroll
    for (int kh = 0; kh < 3; ++kh)
#pragma unroll
        for (int kw = 0; kw < 3; ++kw) {
            int hh = h + kh - 1, ww2 = w + kw - 1;
            if (hh >= 0 && hh < HH && ww2 >= 0 && ww2 < WW)
                s += xc_lin[((size_t)b * LL + hh * WW + ww2) * DIc + c] * cw[c * 9 + kh * 3 + kw];
        }
    xc_act[((size_t)b * LL + rem) * DIc + c] = siluf(s);
}

// ----------------------- 3) global context mean ----------------------------
__global__ void k_gcmean(const float* __restrict__ xc_act, float* __restrict__ gcm)
{
    int b = blockIdx.x, d = threadIdx.x;
    float s = 0.f;
    for (int p = 0; p < LL; ++p)
        s += xc_act[((size_t)b * LL + p) * DIc + d];
    gcm[b * DIc + d] = s * (1.f / LL);
}

// ----------------------- 4) gate = sigmoid(fc2(gelu(fc1(gc)))) -------------
__global__ void k_gate(const float* __restrict__ gcm, const float* __restrict__ fw1,
                       const float* __restrict__ fw2, float* __restrict__ gate)
{
    __shared__ float g[DIc];
    __shared__ float h1[DMm / 4];
    int b = blockIdx.x, d = threadIdx.x;
    g[d] = gcm[b * DIc + d];
    __syncthreads();
    if (d < DMm / 4) {
        float s = 0.f;
        for (int j = 0; j < DMm; ++j) s += fw1[d * DMm + j] * g[j];
        h1[d] = geluf(s);
    }
    __syncthreads();
    float s2 = 0.f;
#pragma unroll
    for (int j = 0; j < DMm / 4; ++j) s2 += fw2[d * (DMm / 4) + j] * h1[j];
    gate[b * DMm + d] = sigm(s2);
}

// ----------------------- 5) build 4 traversal sequences --------------------
// xs layout (K,B,L,D) with D contiguous -> coalesced for GEMM and scan.
__global__ void k_build_xs(const float* __restrict__ xc_act, float* __restrict__ xs)
{
    int l = blockIdx.x, b = blockIdx.y, k = blockIdx.z;
    int d = threadIdx.x;
    int lb = (k < 2) ? l : (LL - 1 - l);                 // k>=2 are flips
    int w_in = lb & 3, h_in = (lb >> 2) & 3;
    int nw = (lb >> 4) % (WW / 4);
    int nh = (lb >> 4) / (WW / 4);
    int p = nh * 4 + h_in, q = nw * 4 + w_in;
    int h = ((k & 1) == 0) ? p : q;                      // k&1 -> transposed scan
    int w = ((k & 1) == 0) ? q : p;
    xs[(((size_t)(k * BB + b)) * LL + l) * DIc + d] =
        xc_act[((size_t)b * LL + h * WW + w) * DIc + d];
}

// ----------------------- 6) x_proj: x_dbl = xs @ x_proj_w^T ----------------
// Uses zero-padded weights (CPAD rows) -> no load guards; store guard only.
__global__ void k_xproj(const float* __restrict__ xs, const float* __restrict__ wpad,
                        float* __restrict__ xdbl)
{
    int k = blockIdx.z;
    const float* X = xs + (size_t)k * BB * LL * DIc;
    const float* W = wpad + (size_t)k * CPAD * DIc;
    float* O = xdbl + (size_t)k * BB * LL * CPROJ;
    const int row0 = blockIdx.y * 16, col0 = blockIdx.x * 32;
    v8f acc[2];
    wmma_xwT_multi<2>(X, W, row0, col0, DIc, DIc, DIc, acc);
    const int lane = threadIdx.x & 31;
    const int nn = lane & 15, mh = (lane >> 4) * 8;
#pragma unroll
    for (int t = 0; t < 2; ++t)
#pragma unroll
        for (int v = 0; v < 8; ++v) {
            int r = row0 + mh + v, c = col0 + t * 16 + nn;
            if (c < CPROJ) O[(size_t)r * CPROJ + c] = acc[t][v];
        }
}

// ------------- 7) dt-proj + bias + softplus -> delta (K,B,L,D) -------------
__global__ void k_dtproj(const float* __restrict__ xdbl, const float* __restrict__ dtw,
                         const float* __restrict__ dtb, float* __restrict__ delta)
{
    int k = blockIdx.z;
    const float* X = xdbl + (size_t)k * BB * LL * CPROJ;   // first RR cols are dts
    const float* W = dtw + (size_t)k * DIc * RR;
    float* O = delta + (size_t)k * BB * LL * DIc;
    const int row0 = blockIdx.y * 16, col0 = blockIdx.x * 32;
    v8f acc[2];
    wmma_xwT_multi<2>(X, W, row0, col0, RR, CPROJ, RR, acc);
    const int lane = threadIdx.x & 31;
    const int nn = lane & 15, mh = (lane >> 4) * 8;
#pragma unroll
    for (int t = 0; t < 2; ++t)
#pragma unroll
        for (int v = 0; v < 8; ++v) {
            int r = row0 + mh + v, c = col0 + t * 16 + nn;
            O[(size_t)r * DIc + c] = softplusf_(acc[t][v] + dtb[k * DIc + c]);
        }
}

// ----------------------- 8) selective scan ---------------------------------
// 16 blocks (one per (b,k) chain), 192 threads = d. Each thread holds h[16],A[16]
// in VGPRs; delta/u coalesced across lanes, B/C broadcast (same-addr, wave dedup).
__global__ void k_scan(const float* __restrict__ xs, const float* __restrict__ xdbl,
                       const float* __restrict__ delta, const float* __restrict__ A_logs,
                       const float* __restrict__ Ds, float* __restrict__ yarr)
{
    int kb = blockIdx.x;
    int k = kb & 3, b = kb >> 2;
    int d = threadIdx.x;
    float A[NN];
#pragma unroll
    for (int n = 0; n < NN; ++n)
        A[n] = -__expf(A_logs[((size_t)(k * DIc + d)) * NN + n]);
    float Dv = Ds[k * DIc + d];
    float h[NN];
#pragma unroll
    for (int n = 0; n < NN; ++n) h[n] = 0.f;
    const size_t seqbase = ((size_t)(k * BB + b)) * LL;
    const float* up = xs    + seqbase * DIc + d;
    const float* dp = delta + seqbase * DIc + d;
    const float* bc = xdbl  + seqbase * CPROJ;
    float* yp       = yarr  + seqbase * DIc + d;
    for (int l = 0; l < LL; ++l) {
        float dt = dp[(size_t)l * DIc];
        float u  = up[(size_t)l * DIc];
        float du = dt * u;
        const float* bcl = bc + (size_t)l * CPROJ;
        float y = 0.f;
#pragma unroll
        for (int n = 0; n < NN; ++n) {
            float Bt = bcl[RR + n];
            float Ct = bcl[RR + NN + n];
            h[n] = __expf(dt * A[n]) * h[n] + du * Bt;
            y += h[n] * Ct;
        }
        yp[(size_t)l * DIc] = y + Dv * u;
    }
}

// --------- 9) combine 4 directions + gate + LayerNorm + multiply z ---------
__global__ void k_combine(const float* __restrict__ yarr, const float* __restrict__ gate,
                          const float* __restrict__ g1, const float* __restrict__ g2,
                          const float* __restrict__ g3, const float* __restrict__ g4,
                          const float* __restrict__ nw, const float* __restrict__ nb,
                          const float* __restrict__ z_silu, float* __restrict__ yfinal)
{
    __shared__ float s1[256], s2[256];
    int lr = blockIdx.x, b = blockIdx.y, d = threadIdx.x;
    int h = lr / WW, w = lr % WW;
    // spatial pixel -> windowed-sequence position
    int l = (((h >> 2) * (WW / 4) + (w >> 2)) << 4) + ((h & 3) << 2) + (w & 3);
    int lt = (l % WW) * HH + (l / WW);                   // wh transpose in seq-space
    const size_t bstride = (size_t)LL * DIc;
    const float* Y0 = yarr + ((size_t)(0 * BB + b)) * bstride;
    const float* Y1 = yarr + ((size_t)(1 * BB + b)) * bstride;
    const float* Y2 = yarr + ((size_t)(2 * BB + b)) * bstride;
    const float* Y3 = yarr + ((size_t)(3 * BB + b)) * bstride;
    float yc = g1[d] * Y0[(size_t)l * DIc + d]
             + g2[d] * Y2[(size_t)(LL - 1 - l) * DIc + d]
             + g3[d] * Y1[(size_t)lt * DIc + d]
             + g4[d] * Y3[(size_t)(LL - 1 - lt) * DIc + d];
    yc *= gate[b * DIc + d];
    s1[d] = yc; s2[d] = yc * yc;
    if (d < 64) { s1[192 + d] = 0.f; s2[192 + d] = 0.f; }
    __syncthreads();
    for (int off = 128; off > 0; off >>= 1) {
        if (d < off) { s1[d] += s1[d + off]; s2[d] += s2[d + off]; }
        __syncthreads();
    }
    float mu  = s1[0] * (1.f / DIc);
    float var = s2[0] * (1.f / DIc) - mu * mu;
    float inv = rsqrtf(var + 1e-5f);
    yc = (yc - mu) * inv * nw[d] + nb[d];
    yc *= z_silu[((size_t)b * LL + lr) * DIc + d];
    yfinal[((size_t)b * LL + lr) * DIc + d] = yc;
}

// ----------------------- 10) out_proj --------------------------------------
__global__ void k_outproj(const float* __restrict__ yf, const float* __restrict__ w,
                          float* __restrict__ out)
{
    const int row0 = blockIdx.y * 16, col0 = blockIdx.x * 32;
    v8f acc[2];
    wmma_xwT_multi<2>(yf, w, row0, col0, DIc, DIc, DIc, acc);
    const int lane = threadIdx.x & 31;
    const int nn = lane & 15, mh = (lane >> 4) * 8;
#pragma unroll
    for (int t = 0; t < 2; ++t)
#pragma unroll
        for (int v = 0; v < 8; ++v) {
            int r = row0 + mh + v, c = col0 + t * 16 + nn;
            out[(size_t)r * DMm + c] = acc[t][v];
        }
}

// ---------------------------------------------------------------------------
extern "C" void kernel_launch(void* const* d_in, const int* in_sizes, int n_in,
                              void* d_out, int out_size, void* d_ws, size_t ws_size,
                              hipStream_t stream)
{
    (void)in_sizes; (void)n_in; (void)out_size; (void)ws_size;
    const float* x          = (const float*)d_in[0];
    const float* in_proj_w  = (const float*)d_in[1];
    const float* conv_w     = (const float*)d_in[2];
    const float* conv_b     = (const float*)d_in[3];
    const float* x_proj_w   = (const float*)d_in[4];
    const float* dt_w       = (const float*)d_in[5];
    const float* dt_b       = (const float*)d_in[6];
    const float* A_logs     = (const float*)d_in[7];
    const float* Ds         = (const float*)d_in[8];
    const float* gamma1     = (const float*)d_in[9];
    const float* gamma2     = (const float*)d_in[10];
    const float* gamma3     = (const float*)d_in[11];
    const float* gamma4     = (const float*)d_in[12];
    const float* fc_w1      = (const float*)d_in[13];
    const float* fc_w2      = (const float*)d_in[14];
    const float* norm_w     = (const float*)d_in[15];
    const float* norm_b     = (const float*)d_in[16];
    const float* out_proj_w = (const float*)d_in[17];
    float* out = (float*)d_out;

    float* ws = (float*)d_ws;
    size_t o = 0;
    float* xc_lin = ws + o; o += (size_t)BB * LL * DIc;          // 7 MB
    float* z_silu = ws + o; o += (size_t)BB * LL * DIc;          // 7 MB
    float* xc_act = ws + o; o += (size_t)BB * LL * DIc;          // 7 MB
    float* xs     = ws + o; o += (size_t)KK * BB * LL * DIc;     // 28 MB
    float* xdbl   = ws + o; o += (size_t)KK * BB * LL * CPROJ;   // 6.5 MB
    float* delta  = ws + o; o += (size_t)KK * BB * LL * DIc;     // 28 MB
    float* yarr   = ws + o; o += (size_t)KK * BB * LL * DIc;     // 28 MB
    float* yfinal = ws + o; o += (size_t)BB * LL * DIc;          // 7 MB
    float* wpad   = ws + o; o += (size_t)KK * CPAD * DIc;        // 192 KB
    float* gcm    = ws + o; o += BB * DIc;
    float* gate   = ws + o; o += BB * DMm;

    k_padw    <<<dim3(KK, CPAD),       DIc, 0, stream>>>(x_proj_w, wpad);
    k_inproj  <<<dim3(12, 576),        32,  0, stream>>>(x, in_proj_w, xc_lin, z_silu);
    k_dwconv  <<<dim3(BB * LL),        DIc, 0, stream>>>(xc_lin, conv_w, conv_b, xc_act);
    k_gcmean  <<<dim3(BB),             DIc, 0, stream>>>(xc_act, gcm);
    k_gate    <<<dim3(BB),             DIc, 0, stream>>>(gcm, fc_w1, fc_w2, gate);
    k_build_xs<<<dim3(LL, BB, KK),     DIc, 0, stream>>>(xc_act, xs);
    k_xproj   <<<dim3(2, 576, KK),     32,  0, stream>>>(xs, wpad, xdbl);
    k_dtproj  <<<dim3(6, 576, KK),     32,  0, stream>>>(xdbl, dt_w, dt_b, delta);
    k_scan    <<<dim3(BB * KK),        DIc, 0, stream>>>(xs, xdbl, delta, A_logs, Ds, yarr);
    k_combine <<<dim3(LL, BB),         DIc, 0, stream>>>(yarr, gate, gamma1, gamma2, gamma3,
                                                         gamma4, norm_w, norm_b, z_silu, yfinal);
    k_outproj <<<dim3(6, 576),         32,  0, stream>>>(yfinal, out_proj_w, out);
}